// LNNFriction_87110526697624
// MI455X (gfx1250) — compile-verified
//
#include <hip/hip_runtime.h>
#include <math.h>

typedef __attribute__((ext_vector_type(2))) float v2f;
typedef __attribute__((ext_vector_type(8))) float v8f;

namespace {
constexpr int kWaves     = 8;     // waves per workgroup
constexpr int kTile      = 16;    // samples per wave
constexpr int kXP        = 68;    // padded activation row pitch (floats) -> conflict-free b64 A-frag loads
constexpr int kChanPitch = kTile * kXP;    // 1088 floats per channel
constexpr int kXWave     = 5 * kChanPitch; // 5440 floats per wave

// shared memory layout (float offsets)
constexpr int OFF_LW1 = 0;                 // lag W1  [64][64]
constexpr int OFF_LW2 = 4096;              // lag W2
constexpr int OFF_FW1 = 8192;              // fric W1
constexpr int OFF_FW2 = 12288;             // fric W2
constexpr int OFF_S   = 16384;
constexpr int OFF_LW0 = OFF_S + 0;         // [2][64]
constexpr int OFF_LB0 = OFF_S + 128;
constexpr int OFF_LB1 = OFF_S + 192;
constexpr int OFF_LB2 = OFF_S + 256;
constexpr int OFF_LW3 = OFF_S + 320;       // [64]
constexpr int OFF_FW0 = OFF_S + 384;
constexpr int OFF_FB0 = OFF_S + 512;
constexpr int OFF_FB1 = OFF_S + 576;
constexpr int OFF_FB2 = OFF_S + 640;
constexpr int OFF_FW3 = OFF_S + 704;
constexpr int OFF_FB3 = OFF_S + 768;       // scalar
constexpr int OFF_X   = OFF_S + 1024;      // per-wave activation tiles
constexpr int kSmemFloats = OFF_X + kWaves * kXWave;   // 60928 floats = 243712 B
}

__device__ __forceinline__ float softplusf_(float x) {
  return (x > 20.0f) ? x : log1pf(expf(x));
}
__device__ __forceinline__ float sigmoidf_(float x) {
  return 1.0f / (1.0f + expf(-x));
}

// One 64->64 lag layer: 5-channel batched GEMM [16,64]x[64,64] via f32 WMMA,
// then softplus chain rule across channels, written back in-place.
__device__ __forceinline__ void lag_layer(float* Xw, const float* W, const float* bvec, int lane) {
  const int mrow = lane & 15;
  const int half = lane >> 4;
  const v8f vzero = {0.f, 0.f, 0.f, 0.f, 0.f, 0.f, 0.f, 0.f};
  v8f acc[5][4];
#pragma unroll
  for (int c = 0; c < 5; ++c)
#pragma unroll
    for (int t = 0; t < 4; ++t) acc[c][t] = vzero;

  for (int k = 0; k < 16; ++k) {
    v2f a[5];
#pragma unroll
    for (int c = 0; c < 5; ++c)
      a[c] = *reinterpret_cast<const v2f*>(&Xw[c * kChanPitch + mrow * kXP + 4 * k + 2 * half]);
    const int row = 4 * k + 2 * half;
#pragma unroll
    for (int t = 0; t < 4; ++t) {
      const int col = t * 16 + mrow;
      v2f b;
      b.x = W[row * 64 + col];
      b.y = W[(row + 1) * 64 + col];
#pragma unroll
      for (int c = 0; c < 5; ++c)
        acc[c][t] = __builtin_amdgcn_wmma_f32_16x16x4_f32(
            false, a[c], false, b, (short)0, acc[c][t], false, false);
    }
  }
  // softplus chain rule + in-place writeback (GEMM fully consumed X above)
#pragma unroll
  for (int t = 0; t < 4; ++t) {
    const int n = t * 16 + mrow;
    const float bias = bvec[n];
#pragma unroll
    for (int r = 0; r < 8; ++r) {
      const int M = r + 8 * half;
      const float v     = acc[0][t][r] + bias;
      const float gq    = acc[1][t][r];
      const float gqd   = acc[2][t][r];
      const float hqqd  = acc[3][t][r];
      const float hqdqd = acc[4][t][r];
      const float s1 = sigmoidf_(v);
      const float s2 = s1 * (1.0f - s1);
      Xw[0 * kChanPitch + M * kXP + n] = softplusf_(v);
      Xw[1 * kChanPitch + M * kXP + n] = s1 * gq;
      Xw[2 * kChanPitch + M * kXP + n] = s1 * gqd;
      Xw[3 * kChanPitch + M * kXP + n] = s1 * hqqd + s2 * gq * gqd;
      Xw[4 * kChanPitch + M * kXP + n] = s1 * hqdqd + s2 * gqd * gqd;
    }
  }
}

// One 64->64 friction layer: 1-channel GEMM + tanh, in-place.
__device__ __forceinline__ void fric_layer(float* X0, const float* W, const float* bvec, int lane) {
  const int mrow = lane & 15;
  const int half = lane >> 4;
  const v8f vzero = {0.f, 0.f, 0.f, 0.f, 0.f, 0.f, 0.f, 0.f};
  v8f acc[4];
#pragma unroll
  for (int t = 0; t < 4; ++t) acc[t] = vzero;

  for (int k = 0; k < 16; ++k) {
    const v2f a = *reinterpret_cast<const v2f*>(&X0[mrow * kXP + 4 * k + 2 * half]);
    const int row = 4 * k + 2 * half;
#pragma unroll
    for (int t = 0; t < 4; ++t) {
      const int col = t * 16 + mrow;
      v2f b;
      b.x = W[row * 64 + col];
      b.y = W[(row + 1) * 64 + col];
      acc[t] = __builtin_amdgcn_wmma_f32_16x16x4_f32(
          false, a, false, b, (short)0, acc[t], false, false);
    }
  }
#pragma unroll
  for (int t = 0; t < 4; ++t) {
    const int n = t * 16 + mrow;
    const float bias = bvec[n];
#pragma unroll
    for (int r = 0; r < 8; ++r) {
      const int M = r + 8 * half;
      X0[M * kXP + n] = tanhf(acc[t][r] + bias);
    }
  }
}

__global__ __launch_bounds__(256, 1) void LNNFriction_87110526697624_kernel(
    const float* __restrict__ lagW0, const float* __restrict__ lagB0,
    const float* __restrict__ lagW1, const float* __restrict__ lagB1,
    const float* __restrict__ lagW2, const float* __restrict__ lagB2,
    const float* __restrict__ lagW3,
    const float* __restrict__ fricW0, const float* __restrict__ fricB0,
    const float* __restrict__ fricW1, const float* __restrict__ fricB1,
    const float* __restrict__ fricW2, const float* __restrict__ fricB2,
    const float* __restrict__ fricW3, const float* __restrict__ fricB3,
    const float* __restrict__ q, const float* __restrict__ qdot,
    float* __restrict__ out, int B) {
  extern __shared__ float smem[];
  const int tid = threadIdx.x;

  // ---- cooperative weight staging into LDS ----
  for (int i = tid; i < 4096; i += 256) {
    smem[OFF_LW1 + i] = lagW1[i];
    smem[OFF_LW2 + i] = lagW2[i];
    smem[OFF_FW1 + i] = fricW1[i];
    smem[OFF_FW2 + i] = fricW2[i];
  }
  if (tid < 128) {
    smem[OFF_LW0 + tid] = lagW0[tid];
    smem[OFF_FW0 + tid] = fricW0[tid];
  }
  if (tid < 64) {
    smem[OFF_LB0 + tid] = lagB0[tid];
    smem[OFF_LB1 + tid] = lagB1[tid];
    smem[OFF_LB2 + tid] = lagB2[tid];
    smem[OFF_LW3 + tid] = lagW3[tid];
    smem[OFF_FB0 + tid] = fricB0[tid];
    smem[OFF_FB1 + tid] = fricB1[tid];
    smem[OFF_FB2 + tid] = fricB2[tid];
    smem[OFF_FW3 + tid] = fricW3[tid];
  }
  if (tid == 0) smem[OFF_FB3] = fricB3[0];
  __syncthreads();

  const int wave = tid >> 5;
  const int lane = tid & 31;
  const int mrow = lane & 15;
  const int half = lane >> 4;
  float* Xw = smem + OFF_X + wave * kXWave;

  const int tb = (blockIdx.x * kWaves + wave) * kTile;
  int gidx = tb + mrow;
  if (gidx >= B) gidx = B - 1;   // clamp (B is a multiple of 128 in practice)
  const float qm  = q[gidx];
  const float qdm = qdot[gidx];

  // ---- Lagrangian net, layer 0 (2 -> 64), 5 channels ----
  for (int jj = 0; jj < 32; ++jj) {
    const int j = 2 * jj + half;
    const float w0  = smem[OFF_LW0 + j];
    const float w1  = smem[OFF_LW0 + 64 + j];
    const float pre = qm * w0 + qdm * w1 + smem[OFF_LB0 + j];
    const float s1 = sigmoidf_(pre);
    const float s2 = s1 * (1.0f - s1);
    Xw[0 * kChanPitch + mrow * kXP + j] = softplusf_(pre);
    Xw[1 * kChanPitch + mrow * kXP + j] = s1 * w0;
    Xw[2 * kChanPitch + mrow * kXP + j] = s1 * w1;
    Xw[3 * kChanPitch + mrow * kXP + j] = s2 * w0 * w1;
    Xw[4 * kChanPitch + mrow * kXP + j] = s2 * w1 * w1;
  }

  // ---- Lagrangian net, layers 1 & 2 (WMMA) ----
  lag_layer(Xw, smem + OFF_LW1, smem + OFF_LB1, lane);
  lag_layer(Xw, smem + OFF_LW2, smem + OFF_LB2, lane);

  // ---- Lagrangian head (64 -> 1): only derivative channels matter ----
  float sg = 0.f, sc = 0.f, sM = 0.f;
  {
    const float* W3 = smem + OFF_LW3;
    const float* Xg = Xw + 1 * kChanPitch + mrow * kXP;
    const float* Xc = Xw + 3 * kChanPitch + mrow * kXP;
    const float* Xm = Xw + 4 * kChanPitch + mrow * kXP;
    for (int jj = 0; jj < 32; ++jj) {
      const int j = half * 32 + jj;
      const float w = W3[j];
      sg += w * Xg[j];
      sc += w * Xc[j];
      sM += w * Xm[j];
    }
    sg += __shfl_xor(sg, 16, 32);
    sc += __shfl_xor(sc, 16, 32);
    sM += __shfl_xor(sM, 16, 32);
  }

  // ---- Friction net (reuses channel-0 region; lag X no longer needed) ----
  for (int jj = 0; jj < 32; ++jj) {
    const int j = 2 * jj + half;
    const float pre = qm * smem[OFF_FW0 + j] + qdm * smem[OFF_FW0 + 64 + j] + smem[OFF_FB0 + j];
    Xw[mrow * kXP + j] = tanhf(pre);
  }
  fric_layer(Xw, smem + OFF_FW1, smem + OFF_FB1, lane);
  fric_layer(Xw, smem + OFF_FW2, smem + OFF_FB2, lane);

  float Ff = 0.f;
  {
    const float* W3 = smem + OFF_FW3;
    const float* X0 = Xw + mrow * kXP;
    for (int jj = 0; jj < 32; ++jj) {
      const int j = half * 32 + jj;
      Ff += W3[j] * X0[j];
    }
    Ff += __shfl_xor(Ff, 16, 32);
    Ff += smem[OFF_FB3];
  }

  // out = (dL/dq - (d2L/dq dqd)*qdot + F) / (d2L/dqd2 + EPS)
  const float res = (sg - sc * qdm + Ff) / (sM + 0.01f);
  if (half == 0 && tb + mrow < B) out[tb + mrow] = res;
}

extern "C" void kernel_launch(void* const* d_in, const int* in_sizes, int n_in,
                              void* d_out, int out_size, void* d_ws, size_t ws_size,
                              hipStream_t stream) {
  (void)n_in; (void)d_ws; (void)ws_size; (void)out_size;
  const float* lagW0  = (const float*)d_in[0];
  const float* lagB0  = (const float*)d_in[1];
  const float* lagW1  = (const float*)d_in[2];
  const float* lagB1  = (const float*)d_in[3];
  const float* lagW2  = (const float*)d_in[4];
  const float* lagB2  = (const float*)d_in[5];
  const float* lagW3  = (const float*)d_in[6];
  // d_in[7] = lag_b3: unused (constant term vanishes under differentiation)
  const float* fricW0 = (const float*)d_in[8];
  const float* fricB0 = (const float*)d_in[9];
  const float* fricW1 = (const float*)d_in[10];
  const float* fricB1 = (const float*)d_in[11];
  const float* fricW2 = (const float*)d_in[12];
  const float* fricB2 = (const float*)d_in[13];
  const float* fricW3 = (const float*)d_in[14];
  const float* fricB3 = (const float*)d_in[15];
  const float* q      = (const float*)d_in[16];
  const float* qdot   = (const float*)d_in[17];

  const int B = in_sizes[16];
  const int samplesPerBlock = kWaves * kTile;  // 128
  const int blocks = (B + samplesPerBlock - 1) / samplesPerBlock;
  const size_t smemBytes = (size_t)kSmemFloats * sizeof(float);

  LNNFriction_87110526697624_kernel<<<blocks, kWaves * 32, smemBytes, stream>>>(
      lagW0, lagB0, lagW1, lagB1, lagW2, lagB2, lagW3,
      fricW0, fricB0, fricW1, fricB1, fricW2, fricB2, fricW3, fricB3,
      q, qdot, (float*)d_out, B);
}